// BaseLayerGate_2834678415368
// MI455X (gfx1250) — compile-verified
//
#include <hip/hip_runtime.h>
#include <hip/hip_bf16.h>
#include <math.h>

// ---------------------------------------------------------------------------
// BaseLayer balanced gating for MI455X (gfx1250, wave32).
//   Phase 1: aff = feats[16384,1024] x centroids^T[1024,128]  via
//            V_WMMA_F32_16X16X4_F32 (fp32 matrix cores, preserves reference
//            numerics), stored column-major s[c][n], c = k*64+e.
//   Phase 2: Sinkhorn reduced to dual potentials u[n],v[e]:
//            la == s - u - v  at all times, so 10 iterations are just
//            u = LSE_e(s - v); v = LSE_n(s - u). s stays L2-resident (8MB).
//   Phase 3: per-column top-256 (iterative argmax, jax tie-break: lower idx),
//            values gathered from original s.
// ---------------------------------------------------------------------------

typedef __attribute__((ext_vector_type(2))) float v2f;
typedef __attribute__((ext_vector_type(8))) float v8f;

#define NTOK   16384   // 4*4096 tokens
#define DMODEL 1024
#define NEXP   64
#define TOPK   2
#define NCOL   (TOPK * NEXP)   // 128
#define CAP    (NTOK / NEXP)   // 256

// ---------------------------------------------------------------------------
// GEMM: each wave owns a 16-row block and all 128 output columns
// (8 accumulator tiles of 16x16 = 64 acc VGPRs/lane). K-loop steps by 4.
// Centroids (512KB) are reused by every wave -> cache resident.
// ---------------------------------------------------------------------------
__global__ __launch_bounds__(128) void gate_gemm_wmma(
    const float* __restrict__ feats,    // [NTOK, DMODEL] row-major
    const float* __restrict__ cent,     // [NCOL, DMODEL] row-major (k,e flat)
    float* __restrict__ s)              // [NCOL, NTOK] column-major scores
{
  const int lane  = threadIdx.x & 31;
  const int wave  = threadIdx.x >> 5;
  const int tileM = (blockIdx.x * 4 + wave) * 16;
  const int m     = lane & 15;          // row within A tile / col within B tile
  const int kb    = (lane >> 4) * 2;    // K sub-offset: lanes 0-15 -> K0,K1; 16-31 -> K2,K3

  v8f acc[8];
#pragma unroll
  for (int t = 0; t < 8; ++t)
#pragma unroll
    for (int i = 0; i < 8; ++i) acc[t][i] = 0.0f;

  const float* arow = feats + (size_t)(tileM + m) * DMODEL + kb;

  for (int k0 = 0; k0 < DMODEL; k0 += 4) {
    v2f a;                               // A fragment: 16x4 fp32 (ISA 7.12.2)
    a.x = arow[k0 + 0];
    a.y = arow[k0 + 1];
#pragma unroll
    for (int ct = 0; ct < 8; ++ct) {     // 8 column tiles of 16 -> 128 cols
      const float* bcol = cent + (size_t)(ct * 16 + m) * DMODEL + k0 + kb;
      v2f b;                             // B fragment: 4x16 fp32
      b.x = bcol[0];
      b.y = bcol[1];
      acc[ct] = __builtin_amdgcn_wmma_f32_16x16x4_f32(
          /*neg_a=*/false, a, /*neg_b=*/false, b,
          /*c_mod=*/(short)0, acc[ct], /*reuse_a=*/false, /*reuse_b=*/false);
    }
  }

  // D layout: VGPR r -> lanes 0-15: (M=r, N=lane); lanes 16-31: (M=r+8, N=lane-16)
  const int rowoff = tileM + ((lane >> 4) ? 8 : 0);
#pragma unroll
  for (int ct = 0; ct < 8; ++ct) {
    const int c = ct * 16 + m;
#pragma unroll
    for (int r = 0; r < 8; ++r)
      s[(size_t)c * NTOK + rowoff + r] = acc[ct][r];
  }
}

__global__ void gate_init_v(float* __restrict__ v)
{
  if (threadIdx.x < NCOL) v[threadIdx.x] = 0.0f;
}

// u[k*NTOK+n] = LSE over e of (s[c][n] - v[c]).  Coalesced: consecutive
// threads hit consecutive n within a column.
__global__ __launch_bounds__(256) void gate_row_lse(
    const float* __restrict__ s, const float* __restrict__ v,
    float* __restrict__ u)
{
  const int t = blockIdx.x * 256 + threadIdx.x;
  if (t >= TOPK * NTOK) return;
  const int k = t >> 14;
  const int n = t & (NTOK - 1);
  float m = -INFINITY, sum = 0.0f;
  for (int e = 0; e < NEXP; ++e) {
    const int c = k * NEXP + e;
    const float x = s[(size_t)c * NTOK + n] - v[c];
    if (x > m) { sum = sum * expf(m - x) + 1.0f; m = x; }
    else       { sum += expf(x - m); }
  }
  u[t] = m + logf(sum);
}

// v[c] = LSE over n of (s[c][n] - u[k][n]).  One workgroup per column.
__global__ __launch_bounds__(256) void gate_col_lse(
    const float* __restrict__ s, const float* __restrict__ u,
    float* __restrict__ v)
{
  __shared__ float sm[256];
  __shared__ float ss[256];
  const int c = blockIdx.x;
  const float* col = s + (size_t)c * NTOK;
  const float* uk  = u + (size_t)(c >> 6) * NTOK;

  float m = -INFINITY, sum = 0.0f;
  for (int n = threadIdx.x; n < NTOK; n += 256) {
    const float x = col[n] - uk[n];
    if (x > m) { sum = sum * expf(m - x) + 1.0f; m = x; }
    else       { sum += expf(x - m); }
  }
  sm[threadIdx.x] = m; ss[threadIdx.x] = sum;
  __syncthreads();
  for (int off = 128; off > 0; off >>= 1) {
    if (threadIdx.x < off) {
      const float m1 = sm[threadIdx.x],        s1 = ss[threadIdx.x];
      const float m2 = sm[threadIdx.x + off],  s2 = ss[threadIdx.x + off];
      const float M  = fmaxf(m1, m2);
      float S = 0.0f;
      if (m1 != -INFINITY) S += s1 * expf(m1 - M);
      if (m2 != -INFINITY) S += s2 * expf(m2 - M);
      sm[threadIdx.x] = M; ss[threadIdx.x] = S;
    }
    __syncthreads();
  }
  if (threadIdx.x == 0) v[c] = sm[0] + logf(ss[0]);
}

// Per-column top-CAP selection.  la scratch rebuilt every launch (graph-safe).
// Tie-break: equal value -> lower token index (matches jax.lax.top_k).
__global__ __launch_bounds__(256) void gate_topk(
    const float* __restrict__ s, const float* __restrict__ u,
    float* __restrict__ la,           // [NCOL, NTOK] mutable scratch
    int* __restrict__ idx_out,        // [NCOL, CAP]
    float* __restrict__ val_out)      // [NCOL, CAP]
{
  __shared__ float rv[256];
  __shared__ int   ri[256];
  const int c = blockIdx.x;
  const float* col  = s  + (size_t)c * NTOK;
  float*       lcol = la + (size_t)c * NTOK;
  const float* uk   = u + (size_t)(c >> 6) * NTOK;

  for (int n = threadIdx.x; n < NTOK; n += 256)
    lcol[n] = col[n] - uk[n];                 // v[c] constant: ordering unchanged
  __threadfence();
  __syncthreads();

  for (int r = 0; r < CAP; ++r) {
    float bm = -INFINITY; int bi = 0x7fffffff;
    for (int n = threadIdx.x; n < NTOK; n += 256) {
      const float x = lcol[n];
      if (x > bm || (x == bm && n < bi)) { bm = x; bi = n; }
    }
    rv[threadIdx.x] = bm; ri[threadIdx.x] = bi;
    __syncthreads();
    for (int off = 128; off > 0; off >>= 1) {
      if (threadIdx.x < off) {
        const float xv = rv[threadIdx.x + off];
        const int   xi = ri[threadIdx.x + off];
        if (xv > rv[threadIdx.x] ||
            (xv == rv[threadIdx.x] && xi < ri[threadIdx.x])) {
          rv[threadIdx.x] = xv; ri[threadIdx.x] = xi;
        }
      }
      __syncthreads();
    }
    if (threadIdx.x == 0) {
      const int w = ri[0];
      idx_out[c * CAP + r] = w;               // int32 bit pattern into d_out
      val_out[c * CAP + r] = col[w];          // gather from ORIGINAL scores
      lcol[w] = -INFINITY;                    // remove winner
      __threadfence();
    }
    __syncthreads();
  }
}

extern "C" void kernel_launch(void* const* d_in, const int* in_sizes, int n_in,
                              void* d_out, int out_size, void* d_ws, size_t ws_size,
                              hipStream_t stream) {
  (void)in_sizes; (void)n_in; (void)out_size; (void)ws_size;
  const float* feats = (const float*)d_in[0];   // [4,4096,1024] -> [16384,1024]
  const float* cent  = (const float*)d_in[1];   // [2,64,1024]   -> [128,1024]

  // Workspace layout (floats): s[128*16384] | la[128*16384] | u[2*16384] | v[128]
  float* s  = (float*)d_ws;
  float* la = s  + (size_t)NCOL * NTOK;
  float* u  = la + (size_t)NCOL * NTOK;
  float* v  = u  + (size_t)TOPK * NTOK;

  int*   idx_out = (int*)d_out;                       // first 2*64*256 entries
  float* val_out = (float*)d_out + NCOL * CAP;        // next  2*64*256 entries

  gate_gemm_wmma<<<NTOK / (16 * 4), 128, 0, stream>>>(feats, cent, s);
  gate_init_v<<<1, 128, 0, stream>>>(v);
  for (int it = 0; it < 10; ++it) {
    gate_row_lse<<<(TOPK * NTOK) / 256, 256, 0, stream>>>(s, v, u);
    gate_col_lse<<<NCOL, 256, 0, stream>>>(s, u, v);
  }
  gate_topk<<<NCOL, 256, 0, stream>>>(s, u, la, idx_out, val_out);
}